// DKEncoder_77489799954619
// MI455X (gfx1250) — compile-verified
//
#include <hip/hip_runtime.h>
#include <hip/hip_bf16.h>
#include <math.h>

// Problem constants (from reference)
#define B_   32
#define S_   128
#define E_   24
#define C_   16
#define D_   16
#define KD_  100
#define QD_  768

typedef __attribute__((ext_vector_type(2))) float v2f;
typedef __attribute__((ext_vector_type(8))) float v8f;

__device__ __forceinline__ float lrelu_mask(float r) {
    // reference: att = where(att==0, -10000, att); att = leaky_relu(att) (slope 0.01)
    float x = (r == 0.0f) ? -10000.0f : r;
    return (x >= 0.0f) ? x : 0.01f * x;
}

// ---------------------------------------------------------------------------
// Stage 1: qi[l][b][j] = tanh( sum_k q[b,0,k] * Wq_l[j,k] + bq_l[j] )
// GEMM (32 x 768) x (768 x 100)^T via V_WMMA_F32_16X16X4_F32.
// Grid: 2 layers * 2 M-tiles * 7 N-tiles = 28 waves (1 wave per block).
// Padded N-columns (100..111) handled branch-free: clamped address + 0-mask,
// so the K-loop is unconditional loads + wmma (no exec save/restore).
// ---------------------------------------------------------------------------
__global__ __launch_bounds__(32)
void qi_gemm_kernel(const float* __restrict__ q,
                    const float* __restrict__ Wq2, const float* __restrict__ bq2,
                    const float* __restrict__ Wq1, const float* __restrict__ bq1,
                    float* __restrict__ qi)  // (2,B,KD)
{
    int bx = blockIdx.x;
    int l  = bx / 14;           // layer: 0 -> layer2 params, 1 -> layer1 params
    int mt = (bx / 7) & 1;      // batch tile
    int nt = bx % 7;            // output-column tile (covers 112 >= 100)
    const float* Wq = (l == 0) ? Wq2 : Wq1;
    const float* bq = (l == 0) ? bq2 : bq1;

    int lane = threadIdx.x;
    int half = lane >> 4;       // 0: lanes 0-15, 1: lanes 16-31
    int l16  = lane & 15;

    // A: 16x4 f32 tile of q0. Lane holds row m=l16, K = kbase + 2*half {+0,+1}.
    const float* arow = q + (size_t)(mt * 16 + l16) * (S_ * QD_);   // q[b,0,:]
    // B: 4x16 f32 tile of Wq^T. Lane holds col n=l16, K = kbase + 2*half {+0,+1}.
    int n  = nt * 16 + l16;
    int nc = (n < KD_) ? n : (KD_ - 1);            // clamped (always valid) row
    float bmask = (n < KD_) ? 1.0f : 0.0f;         // zero out padded columns
    const float* brow = Wq + (size_t)nc * QD_;

    v8f acc = {};
    #pragma unroll 4
    for (int k = 0; k < QD_; k += 4) {
        int ka = k + 2 * half;
        v2f A, Bv;
        A.x  = arow[ka];
        A.y  = arow[ka + 1];
        Bv.x = brow[ka]     * bmask;
        Bv.y = brow[ka + 1] * bmask;
        acc = __builtin_amdgcn_wmma_f32_16x16x4_f32(
                  false, A, false, Bv, (short)0, acc, false, false);
    }

    // C/D layout: VGPR r, lane -> (M = r + 8*half, N = l16)
    if (n < KD_) {
        float bias = bq[n];
        #pragma unroll
        for (int r = 0; r < 8; ++r) {
            int bRow = mt * 16 + r + 8 * half;
            qi[(size_t)l * B_ * KD_ + (size_t)bRow * KD_ + n] = tanhf(acc[r] + bias);
        }
    }
}

// ---------------------------------------------------------------------------
// Stage 2: w[l][b][i] = sum_j qi[l][b][j] * Wkv_l[j][i]   ((32x100)x(100x100))
// Same WMMA tiling; K = 100 = 25 steps of 4. Branch-free padding as above.
// ---------------------------------------------------------------------------
__global__ __launch_bounds__(32)
void w_gemm_kernel(const float* __restrict__ qi,
                   const float* __restrict__ Wkv2, const float* __restrict__ Wkv1,
                   float* __restrict__ w)   // (2,B,KD)
{
    int bx = blockIdx.x;
    int l  = bx / 14;
    int mt = (bx / 7) & 1;
    int nt = bx % 7;
    const float* Wkv = (l == 0) ? Wkv2 : Wkv1;

    int lane = threadIdx.x;
    int half = lane >> 4;
    int l16  = lane & 15;

    const float* arow = qi + (size_t)l * B_ * KD_ + (size_t)(mt * 16 + l16) * KD_;
    int n  = nt * 16 + l16;
    int nc = (n < KD_) ? n : (KD_ - 1);
    float bmask = (n < KD_) ? 1.0f : 0.0f;
    const float* bcol = Wkv + nc;                   // column n of Wkv (stride KD_)

    v8f acc = {};
    #pragma unroll 5
    for (int k = 0; k < KD_; k += 4) {      // 100 % 4 == 0
        int ka = k + 2 * half;
        v2f A, Bv;
        A.x  = arow[ka];
        A.y  = arow[ka + 1];
        Bv.x = bcol[(size_t)ka * KD_]       * bmask;
        Bv.y = bcol[(size_t)(ka + 1) * KD_] * bmask;
        acc = __builtin_amdgcn_wmma_f32_16x16x4_f32(
                  false, A, false, Bv, (short)0, acc, false, false);
    }

    if (n < KD_) {
        #pragma unroll
        for (int r = 0; r < 8; ++r) {
            int bRow = mt * 16 + r + 8 * half;
            w[(size_t)l * B_ * KD_ + (size_t)bRow * KD_ + n] = acc[r];
        }
    }
}

// ---------------------------------------------------------------------------
// Stage 3 (dominant, bandwidth-bound): per (b,e):
//   att[c,d]  = dot(w2[b], k2[b,e,c,d,:]) / 10  -> mask/leaky/softmax over d
//   comb2[b,e,c,:] = sum_d att[c,d] * v2[b,e,c,d,:]
// Streams k2 + v2 (157 MB) exactly once, coalesced.
// ---------------------------------------------------------------------------
__global__ __launch_bounds__(256)
void layer2_kernel(const float* __restrict__ k2, const float* __restrict__ v2,
                   const float* __restrict__ w,      // (2,B,KD): l=0 half
                   float* __restrict__ comb2)        // (B,E,C,KD)
{
    int be = blockIdx.x;            // b*E + e
    int b  = be / E_;
    __shared__ float sw[KD_];
    __shared__ float satt[C_ * D_];

    int t = threadIdx.x;
    if (t < KD_) sw[t] = w[(size_t)b * KD_ + t];
    __syncthreads();

    int wave = t >> 5, lane = t & 31;
    const float* k2base = k2 + (size_t)be * (C_ * D_ * KD_);

    // 256 rows of length 100; 8 waves x 32 rows, wave-cooperative dots
    for (int r = wave * 32; r < wave * 32 + 32; ++r) {
        const float* row = k2base + (size_t)r * KD_;
        __builtin_prefetch(row + KD_ + lane * 4, 0, 1);   // next row -> global_prefetch_b8
        float acc = 0.0f;
        for (int i = lane; i < KD_; i += 32) acc += row[i] * sw[i];
        #pragma unroll
        for (int off = 16; off > 0; off >>= 1) acc += __shfl_down(acc, off, 32);
        if (lane == 0) satt[r] = acc * 0.1f;              // 1/sqrt(100)
    }
    __syncthreads();

    // Softmax over d within each c (redundantly per-thread; 16 elems, trivial)
    {
        float raw = satt[t];
        int c = t >> 4;
        float m = -3.4e38f;
        for (int dd = 0; dd < D_; ++dd) m = fmaxf(m, lrelu_mask(satt[c * D_ + dd]));
        float ssum = 0.0f;
        for (int dd = 0; dd < D_; ++dd) ssum += __expf(lrelu_mask(satt[c * D_ + dd]) - m);
        float a = __expf(lrelu_mask(raw) - m) / ssum;
        if (a == 0.0625f) a = 0.0f;                       // where(att == 1/16, 0, att)
        __syncthreads();
        satt[t] = a;
    }
    __syncthreads();

    // Weighted sum over d: wave handles c = wave, wave+8; lanes over kd (coalesced)
    const float* v2base = v2 + (size_t)be * (C_ * D_ * KD_);
    float* outbase = comb2 + (size_t)be * (C_ * KD_);
    for (int c = wave; c < C_; c += 8) {
        for (int kd = lane; kd < KD_; kd += 32) {
            float acc = 0.0f;
            #pragma unroll
            for (int d = 0; d < D_; ++d)
                acc += satt[c * D_ + d] * v2base[(size_t)(c * D_ + d) * KD_ + kd];
            outbase[(size_t)c * KD_ + kd] = acc;
        }
    }
}

// ---------------------------------------------------------------------------
// Stage 4: layer-1 per (b,e): att1[c] = dot(w1[b], k1[b,e,c,:]) / 10;
// softmax over c; out[j] = sum_c att1[c] * vcat[c][j], vcat = [v1 | comb2]
// ---------------------------------------------------------------------------
__global__ __launch_bounds__(128)
void layer1_kernel(const float* __restrict__ k1, const float* __restrict__ v1,
                   const float* __restrict__ w,      // (2,B,KD): l=1 half
                   const float* __restrict__ comb2,  // (B,E,C,KD)
                   float* __restrict__ comb1)        // (B,E,2*KD)
{
    int be = blockIdx.x;
    int b  = be / E_;
    __shared__ float sw[KD_];
    __shared__ float satt[C_];

    int t = threadIdx.x;
    if (t < KD_) sw[t] = w[(size_t)(B_ + b) * KD_ + t];
    __syncthreads();

    int wave = t >> 5, lane = t & 31;       // 4 waves
    const float* k1base = k1 + (size_t)be * (C_ * KD_);
    for (int c = wave; c < C_; c += 4) {
        const float* row = k1base + (size_t)c * KD_;
        float acc = 0.0f;
        for (int i = lane; i < KD_; i += 32) acc += row[i] * sw[i];
        #pragma unroll
        for (int off = 16; off > 0; off >>= 1) acc += __shfl_down(acc, off, 32);
        if (lane == 0) satt[c] = acc * 0.1f;
    }
    __syncthreads();

    float a_val = 0.0f;
    if (t < C_) {
        float m = -3.4e38f;
        for (int cc = 0; cc < C_; ++cc) m = fmaxf(m, lrelu_mask(satt[cc]));
        float ssum = 0.0f;
        for (int cc = 0; cc < C_; ++cc) ssum += __expf(lrelu_mask(satt[cc]) - m);
        a_val = __expf(lrelu_mask(satt[t]) - m) / ssum;
        if (a_val == 0.0625f) a_val = 0.0f;
    }
    __syncthreads();
    if (t < C_) satt[t] = a_val;
    __syncthreads();

    const float* v1base = v1 + (size_t)be * (C_ * KD_);
    const float* c2base = comb2 + (size_t)be * (C_ * KD_);
    float* ob = comb1 + (size_t)be * (2 * KD_);
    for (int j = t; j < 2 * KD_; j += 128) {
        float acc = 0.0f;
        #pragma unroll 4
        for (int c = 0; c < C_; ++c) {
            float vv = (j < KD_) ? v1base[(size_t)c * KD_ + j]
                                 : c2base[(size_t)c * KD_ + (j - KD_)];
            acc += satt[c] * vv;
        }
        ob[j] = acc;
    }
}

// ---------------------------------------------------------------------------
// Stage 5: rank = clip(cumsum(mask)-1, 0, E-1); gather rows of comb1, mask out
// ---------------------------------------------------------------------------
__global__ __launch_bounds__(256)
void gather_kernel(const int* __restrict__ input_ent,   // (B,S) int32
                   const float* __restrict__ comb1,     // (B,E,200)
                   float* __restrict__ out)             // (B,S,200)
{
    int b = blockIdx.x;
    __shared__ int srank[S_];
    __shared__ int smask[S_];
    int t = threadIdx.x;
    if (t == 0) {
        int run = 0;
        for (int s = 0; s < S_; ++s) {
            int m = (input_ent[b * S_ + s] != 0) ? 1 : 0;
            smask[s] = m;
            run += m;
            int r = run - 1;
            if (r < 0) r = 0;
            if (r > E_ - 1) r = E_ - 1;
            srank[s] = r;
        }
    }
    __syncthreads();

    const float* cb = comb1 + (size_t)b * E_ * (2 * KD_);
    float* ob = out + (size_t)b * S_ * (2 * KD_);
    for (int idx = t; idx < S_ * 2 * KD_; idx += 256) {
        int s = idx / (2 * KD_);
        int j = idx - s * (2 * KD_);
        ob[idx] = smask[s] ? cb[(size_t)srank[s] * (2 * KD_) + j] : 0.0f;
    }
}

// ---------------------------------------------------------------------------
extern "C" void kernel_launch(void* const* d_in, const int* in_sizes, int n_in,
                              void* d_out, int out_size, void* d_ws, size_t ws_size,
                              hipStream_t stream) {
    const int*   input_ent = (const int*)  d_in[0];
    const float* q         = (const float*)d_in[1];
    const float* k1        = (const float*)d_in[2];
    const float* v1        = (const float*)d_in[3];
    const float* k2        = (const float*)d_in[4];
    const float* v2        = (const float*)d_in[5];
    const float* Wkv2      = (const float*)d_in[6];
    const float* Wq2       = (const float*)d_in[7];
    const float* bq2       = (const float*)d_in[8];
    const float* Wkv1      = (const float*)d_in[9];
    const float* Wq1       = (const float*)d_in[10];
    const float* bq1       = (const float*)d_in[11];
    float* out = (float*)d_out;

    // Workspace partition (floats)
    float* ws    = (float*)d_ws;
    float* qi    = ws;                         // 2*B*KD     = 6400
    float* w     = ws + 2 * B_ * KD_;          // 2*B*KD     = 6400
    float* comb2 = w  + 2 * B_ * KD_;          // B*E*C*KD   = 1,228,800
    float* comb1 = comb2 + (size_t)B_ * E_ * C_ * KD_;   // B*E*200 = 153,600

    qi_gemm_kernel<<<28, 32, 0, stream>>>(q, Wq2, bq2, Wq1, bq1, qi);
    w_gemm_kernel <<<28, 32, 0, stream>>>(qi, Wkv2, Wkv1, w);
    layer2_kernel <<<B_ * E_, 256, 0, stream>>>(k2, v2, w, comb2);
    layer1_kernel <<<B_ * E_, 128, 0, stream>>>(k1, v1, w, comb2, comb1);
    gather_kernel <<<B_, 256, 0, stream>>>(input_ent, comb1, out);
}